// GraphConvBn_44633300140134
// MI455X (gfx1250) — compile-verified
//
#include <hip/hip_runtime.h>
#include <math.h>

typedef __attribute__((ext_vector_type(2))) float v2f;
typedef __attribute__((ext_vector_type(8))) float v8f;

#define FDIM 64
#define NGRAPH 64

// ---------------- degree / normalization ----------------
__global__ __launch_bounds__(256) void k_deg_init(float* __restrict__ deg, int N) {
  int i = blockIdx.x * 256 + threadIdx.x;
  if (i < N) deg[i] = 1.0f;  // self-loop contributes 1 to every node's degree
}

__global__ __launch_bounds__(256) void k_deg_edges(const int* __restrict__ dst,
                                                   float* __restrict__ deg, int E) {
  int e = blockIdx.x * 256 + threadIdx.x;
  if (e < E) unsafeAtomicAdd(&deg[dst[e]], 1.0f);
}

__global__ __launch_bounds__(256) void k_dinv(const float* __restrict__ deg,
                                              float* __restrict__ dinv, int N) {
  int i = blockIdx.x * 256 + threadIdx.x;
  if (i < N) dinv[i] = rsqrtf(deg[i]);  // deg >= 1 always (self-loop)
}

// ---------------- fp32 WMMA GEMM: h[N,64] = x[N,64] @ W[64,64] ----------------
// One wave computes a 16x64 output tile: 16 K-steps (K=4 each) x 4 N-tiles.
// A layout (16x4 f32, 2 VGPR): lane<16 -> M=lane, k-pair {0,1}; lane>=16 -> M=lane-16, {2,3}
// B layout (4x16 f32, 2 VGPR): lane<16 -> N=lane, k {0,1};     lane>=16 -> N=lane-16, {2,3}
// C/D (16x16 f32, 8 VGPR): vgpr r, lane<16 -> M=r, N=lane; lane>=16 -> M=r+8, N=lane-16
__global__ __launch_bounds__(256) void k_gemm_wmma(const float* __restrict__ x,
                                                   const float* __restrict__ W,
                                                   float* __restrict__ h, int tiles) {
  int wave = blockIdx.x * 8 + (threadIdx.x >> 5);
  int lane = threadIdx.x & 31;
  if (wave >= tiles) return;  // uniform per wave -> EXEC all-ones at WMMA
  int m0    = wave << 4;
  int lrow  = lane & 15;
  int khalf = lane >> 4;  // 0: k pair {0,1}; 1: k pair {2,3}

  v8f acc0 = {}, acc1 = {}, acc2 = {}, acc3 = {};
  const float* xp = x + (size_t)(m0 + lrow) * FDIM + khalf * 2;
  const float* wp = W + (size_t)(khalf * 2) * FDIM + lrow;

#pragma unroll
  for (int kk = 0; kk < 16; ++kk) {
    v2f a;
    a.x = xp[0];
    a.y = xp[1];
    v2f b0, b1, b2, b3;
    b0.x = wp[0];  b0.y = wp[FDIM + 0];
    b1.x = wp[16]; b1.y = wp[FDIM + 16];
    b2.x = wp[32]; b2.y = wp[FDIM + 32];
    b3.x = wp[48]; b3.y = wp[FDIM + 48];
    acc0 = __builtin_amdgcn_wmma_f32_16x16x4_f32(false, a, false, b0, (short)0, acc0, false, false);
    acc1 = __builtin_amdgcn_wmma_f32_16x16x4_f32(false, a, false, b1, (short)0, acc1, false, false);
    acc2 = __builtin_amdgcn_wmma_f32_16x16x4_f32(false, a, false, b2, (short)0, acc2, false, false);
    acc3 = __builtin_amdgcn_wmma_f32_16x16x4_f32(false, a, false, b3, (short)0, acc3, false, false);
    xp += 4;
    wp += 4 * FDIM;
  }

  int rbase = m0 + khalf * 8;
#pragma unroll
  for (int r = 0; r < 8; ++r) {
    float* hp = h + (size_t)(rbase + r) * FDIM + lrow;
    hp[0]  = acc0[r];
    hp[16] = acc1[r];
    hp[32] = acc2[r];
    hp[48] = acc3[r];
  }
}

// scalar fallback for a ragged tail (never fires for N=50000, kept for safety)
__global__ __launch_bounds__(256) void k_gemm_rem(const float* __restrict__ x,
                                                  const float* __restrict__ W,
                                                  float* __restrict__ h, int row0, int N) {
  int t = blockIdx.x * 256 + threadIdx.x;
  int i = row0 + (t >> 6), f = t & 63;
  if (i < N) {
    float s = 0.f;
    for (int k = 0; k < FDIM; ++k) s += x[(size_t)i * FDIM + k] * W[k * FDIM + f];
    h[(size_t)i * FDIM + f] = s;
  }
}

// ---------------- out = h * dinv^2 + b  (self-loop term + bias, float4) ----------------
__global__ __launch_bounds__(256) void k_out_init(const float* __restrict__ h,
                                                  const float* __restrict__ dinv,
                                                  const float* __restrict__ b,
                                                  float* __restrict__ out, int N) {
  int t = blockIdx.x * 256 + threadIdx.x;  // N*16 threads, 4 feats each
  if (t < N * 16) {
    int i = t >> 4, c = (t & 15) * 4;
    float di  = dinv[i];
    float nrm = di * di;
    float4 hv = *(const float4*)(h + (size_t)i * FDIM + c);
    float4 o;
    o.x = hv.x * nrm + b[c + 0];
    o.y = hv.y * nrm + b[c + 1];
    o.z = hv.z * nrm + b[c + 2];
    o.w = hv.w * nrm + b[c + 3];
    *(float4*)(out + (size_t)i * FDIM + c) = o;
  }
}

// ---------------- edge scatter: out[d] += dinv[s]*dinv[d]*h[s] ----------------
__global__ __launch_bounds__(256) void k_scatter(const int* __restrict__ src,
                                                 const int* __restrict__ dst,
                                                 const float* __restrict__ dinv,
                                                 const float* __restrict__ h,
                                                 float* __restrict__ out, int E) {
  long long t = (long long)blockIdx.x * 256 + threadIdx.x;
  int e = (int)(t >> 6), f = (int)(t & 63);
  if (e < E) {
    int s = src[e], d = dst[e];
    float nrm = dinv[s] * dinv[d];
    unsafeAtomicAdd(out + (size_t)d * FDIM + f, nrm * h[(size_t)s * FDIM + f]);
  }
}

// ---------------- per-graph stats: sum, sumsq, count (batch is sorted) ----------------
__global__ __launch_bounds__(256) void k_stats(const float* __restrict__ out,
                                               const int* __restrict__ batch, int N,
                                               float* __restrict__ sum, float* __restrict__ sq,
                                               float* __restrict__ cnt) {
  int g = blockIdx.x;
  int lo = 0, hi = N;
  while (lo < hi) { int mid = (lo + hi) >> 1; if (batch[mid] < g) lo = mid + 1; else hi = mid; }
  int start = lo;
  hi = N;
  while (lo < hi) { int mid = (lo + hi) >> 1; if (batch[mid] < g + 1) lo = mid + 1; else hi = mid; }
  int end = lo;

  int f  = threadIdx.x & 63;
  int rg = threadIdx.x >> 6;  // 4 row groups
  float s = 0.f, q = 0.f;
  for (int i = start + rg; i < end; i += 4) {
    float v = out[(size_t)i * FDIM + f];
    s += v;
    q += v * v;
  }
  __shared__ float ls[4][FDIM];
  __shared__ float lq[4][FDIM];
  ls[rg][f] = s;
  lq[rg][f] = q;
  __syncthreads();
  if (rg == 0) {
    s = ls[0][f] + ls[1][f] + ls[2][f] + ls[3][f];
    q = lq[0][f] + lq[1][f] + lq[2][f] + lq[3][f];
    sum[g * FDIM + f] = s;
    sq[g * FDIM + f]  = q;
    if (f == 0) cnt[g] = (float)(end - start);
  }
}

// ---------------- fused GraphNorm + affine + ReLU (in place on d_out) ----------------
// var = E[out^2] - (2s - s^2) * mean^2   where s = gn_mean_scale
__global__ __launch_bounds__(256) void k_final(float* __restrict__ out,
                                               const int* __restrict__ batch,
                                               const float* __restrict__ sum,
                                               const float* __restrict__ sq,
                                               const float* __restrict__ cnt,
                                               const float* __restrict__ gms,
                                               const float* __restrict__ w,
                                               const float* __restrict__ bias, int N) {
  long long t = (long long)blockIdx.x * 256 + threadIdx.x;
  if (t < (long long)N * FDIM) {
    int i = (int)(t >> 6), f = (int)(t & 63);
    int g = batch[i];
    float c    = cnt[g];
    float mean = sum[g * FDIM + f] / c;
    float msq  = sq[g * FDIM + f] / c;
    float sc   = gms[f];
    float var  = fmaxf(msq - (2.f * sc - sc * sc) * mean * mean, 0.f);
    float v    = (out[t] - sc * mean) * rsqrtf(var + 1e-5f);
    float r    = w[f] * v + bias[f];
    out[t]     = fmaxf(r, 0.f);
  }
}

extern "C" void kernel_launch(void* const* d_in, const int* in_sizes, int n_in,
                              void* d_out, int out_size, void* d_ws, size_t ws_size,
                              hipStream_t stream) {
  const float* x    = (const float*)d_in[0];
  const float* W    = (const float*)d_in[1];
  const float* b    = (const float*)d_in[2];
  const float* gnw  = (const float*)d_in[3];
  const float* gnb  = (const float*)d_in[4];
  const float* gms  = (const float*)d_in[5];
  const int*   ei   = (const int*)d_in[6];
  const int*   batch = (const int*)d_in[7];

  int N = in_sizes[0] / FDIM;
  int E = in_sizes[6] / 2;
  const int* src = ei;
  const int* dst = ei + E;
  float* out = (float*)d_out;

  float* h    = (float*)d_ws;                 // N*64
  float* deg  = h + (size_t)N * FDIM;         // N
  float* dinv = deg + N;                      // N
  float* sum  = dinv + N;                     // 64*64
  float* sq   = sum + NGRAPH * FDIM;          // 64*64
  float* cnt  = sq + NGRAPH * FDIM;           // 64

  k_deg_init<<<(N + 255) / 256, 256, 0, stream>>>(deg, N);
  k_deg_edges<<<(E + 255) / 256, 256, 0, stream>>>(dst, deg, E);
  k_dinv<<<(N + 255) / 256, 256, 0, stream>>>(deg, dinv, N);

  int tiles = N / 16;
  if (tiles > 0) k_gemm_wmma<<<(tiles + 7) / 8, 256, 0, stream>>>(x, W, h, tiles);
  int rem = N - tiles * 16;
  if (rem > 0) k_gemm_rem<<<(rem * 64 + 255) / 256, 256, 0, stream>>>(x, W, h, tiles * 16, N);

  k_out_init<<<((size_t)N * 16 + 255) / 256, 256, 0, stream>>>(h, dinv, b, out, N);

  long long work = (long long)E * FDIM;
  k_scatter<<<(unsigned)((work + 255) / 256), 256, 0, stream>>>(src, dst, dinv, h, out, E);

  k_stats<<<NGRAPH, 256, 0, stream>>>(out, batch, N, sum, sq, cnt);

  long long fin = (long long)N * FDIM;
  k_final<<<(unsigned)((fin + 255) / 256), 256, 0, stream>>>(out, batch, sum, sq, cnt, gms, gnw, gnb, N);
}